// GraphConvolutionLayer_3770981286186
// MI455X (gfx1250) — compile-verified
//
#include <hip/hip_runtime.h>

typedef __attribute__((ext_vector_type(2))) float v2f;
typedef __attribute__((ext_vector_type(8))) float v8f;
typedef __attribute__((ext_vector_type(4))) unsigned int v4u;
typedef __attribute__((ext_vector_type(4))) int v4i;
typedef __attribute__((ext_vector_type(8))) int v8i;

#define N_NODES 50000
#define N_EDGES 800000
#define F 64                      // F_IN == F_OUT == 64
#define ROW_TILES (N_NODES / 16)  // 3125, exact
#define TILES_PER_BLOCK 8

#if __has_builtin(__builtin_amdgcn_tensor_load_to_lds) && \
    __has_builtin(__builtin_amdgcn_s_wait_tensorcnt)
#define USE_TDM 1
#else
#define USE_TDM 0
#endif

// ---------------------------------------------------------------------------
// Kernel 1: out[r][f] = bias[f]   (out must be re-initialized every call)
// ---------------------------------------------------------------------------
__global__ void gcn_init_bias(float* __restrict__ out, const float* __restrict__ bias) {
    int i = blockIdx.x * blockDim.x + threadIdx.x;
    if (i < N_NODES * F) out[i] = bias[i & (F - 1)];
}

// ---------------------------------------------------------------------------
// Kernel 2: H = feature @ weight  via V_WMMA_F32_16X16X4_F32.
// Weight (16 KB) is staged into LDS once per workgroup by the Tensor Data
// Mover (tensor_load_to_lds, TENSORcnt), then B fragments are read from LDS
// as ds_load_2addr pairs.  One wave computes a 16-row x 64-col tile of H.
//   A (16x4 f32):  lane -> m = lane&15; VGPR{0,1} hold K = (lane>>4)*2 + {0,1}
//   B (4x16 f32):  lane -> n = lane&15; VGPR{0,1} hold K rows (lane>>4)*2 + {0,1}
//   C/D (16x16):   lane -> n = lane&15; VGPR i holds row  i + (lane>>4)*8
// ---------------------------------------------------------------------------
__global__ void __launch_bounds__(256)
gcn_gemm_wmma(const float* __restrict__ A,   // feature [N_NODES, 64]
              const float* __restrict__ W,   // weight  [64, 64] (K-major rows)
              float* __restrict__ H) {       // output  [N_NODES, 64]
    __shared__ float Wlds[F * F];            // 16 KB, sole LDS object -> offset 0

#if USE_TDM
    if (threadIdx.x < 32) {                  // one wave issues the TDM op
        const unsigned long long wa = (unsigned long long)(const void*)W;
        // D# group 0: count=1 | lds_addr=0 | global_addr | type=2 ("image")
        v4u g0;
        g0[0] = 1u;                               // count = 1 (valid descriptor)
        g0[1] = 0u;                               // lds_addr = 0
        g0[2] = (unsigned int)wa;                 // global_addr[31:0]
        g0[3] = ((unsigned int)(wa >> 32) & 0x01FFFFFFu) | 0x80000000u;  // [56:32] | type=2
        // D# group 1: data_size=4B; 1-D tensor of 4096 elements, one 4096-elem tile
        v8i g1;
        g1[0] = 0x00020000;                       // data_size = 2 (4 bytes)
        g1[1] = (int)(4096u << 16);               // tensor_dim0[15:0] in [31:16]
        g1[2] = 0x00010000;                       // tensor_dim0 hi = 0; tensor_dim1 = 1
        g1[3] = (int)(4096u << 16);               // tile_dim0 = 4096
        g1[4] = 1;                                // tile_dim1 = 1, tile_dim2 = 0
        g1[5] = 4096;                             // tensor_dim0_stride[31:0]
        g1[6] = (int)(4096u << 16);               // tensor_dim1_stride[15:0]
        g1[7] = 0;
        v4i gz4 = {0, 0, 0, 0};                   // groups 2/3: unused (zeros)
        v8i gz8 = {0, 0, 0, 0, 0, 0, 0, 0};
        __builtin_amdgcn_tensor_load_to_lds(g0, g1, gz4, gz4, gz8, 0);
        __builtin_amdgcn_s_wait_tensorcnt(0);     // LDS data landed for this wave
    }
    __syncthreads();                              // publish to all waves
    // The TDM builtin writes LDS through an integer descriptor the compiler
    // cannot see as aliasing Wlds; without this escape+clobber LLVM folds all
    // Wlds loads to undef (observed round 3: ds=0, garbage B operands).
    asm volatile("" : : "r"(&Wlds[0]) : "memory");
#else
    for (int i = threadIdx.x; i < (F * F) / 4; i += 256)
        ((float4*)Wlds)[i] = ((const float4*)W)[i];
    __syncthreads();
#endif

    const int lane = threadIdx.x & 31;
    const int wave = threadIdx.x >> 5;
    const int tile = blockIdx.x * TILES_PER_BLOCK + wave;  // 16-row tile index

    if (tile < ROW_TILES) {
        const int row0 = tile * 16;
        const int nm   = lane & 15;   // m for A, n for B/C/D
        const int half = lane >> 4;   // 0 or 1

        v8f acc[4] = {v8f{}, v8f{}, v8f{}, v8f{}};

        for (int kk = 0; kk < 16; ++kk) {
            const int k0 = kk * 4 + half * 2;     // this lane's first K

            // A tile: two consecutive f32 -> global_load_b64
            v2f a;
            const float* ap = A + (size_t)(row0 + nm) * F + k0;
            a.x = ap[0];
            a.y = ap[1];

#pragma unroll
            for (int nt = 0; nt < 4; ++nt) {
                // B fragment from LDS: pair 256 B apart -> ds_load_2addr_b32
                const float* wl = Wlds + k0 * F + nt * 16 + nm;
                v2f b;
                b.x = wl[0];
                b.y = wl[F];
                acc[nt] = __builtin_amdgcn_wmma_f32_16x16x4_f32(
                    /*neg_a=*/false, a, /*neg_b=*/false, b,
                    /*c_mod=*/(short)0, acc[nt],
                    /*reuse_a=*/false, /*reuse_b=*/false);
            }
        }

        // Store C/D: acc[nt][i] = H[row0 + i + half*8][nt*16 + nm]
#pragma unroll
        for (int nt = 0; nt < 4; ++nt) {
#pragma unroll
            for (int i = 0; i < 8; ++i) {
                H[(size_t)(row0 + i + half * 8) * F + nt * 16 + nm] = acc[nt][i];
            }
        }
    }
}

// ---------------------------------------------------------------------------
// Kernel 3: COO scatter.  One wave per edge, 2 features per lane.
//   out[r][f] += v * H[c][f]   with hardware f32 atomics (L2-resident).
// ---------------------------------------------------------------------------
__global__ void __launch_bounds__(256)
gcn_scatter(const int* __restrict__ rows, const int* __restrict__ cols,
            const float* __restrict__ vals, const float* __restrict__ H,
            float* out) {
    const int e = blockIdx.x * 8 + (threadIdx.x >> 5);
    if (e >= N_EDGES) return;
    const int lane = threadIdx.x & 31;

    const int r = rows[e];          // wave-uniform -> scalar load
    const int c = cols[e];
    const float v = vals[e];

    const float2 hv = *(const float2*)(H + (size_t)c * F + lane * 2);
    float* op = out + (size_t)r * F + lane * 2;
    unsafeAtomicAdd(op + 0, v * hv.x);
    unsafeAtomicAdd(op + 1, v * hv.y);
}

// ---------------------------------------------------------------------------
extern "C" void kernel_launch(void* const* d_in, const int* in_sizes, int n_in,
                              void* d_out, int out_size, void* d_ws, size_t ws_size,
                              hipStream_t stream) {
    const int*   rows = (const int*)d_in[0];
    const int*   cols = (const int*)d_in[1];
    const float* vals = (const float*)d_in[2];
    const float* feat = (const float*)d_in[3];
    const float* W    = (const float*)d_in[4];
    const float* bias = (const float*)d_in[5];
    float* out = (float*)d_out;
    float* H   = (float*)d_ws;                 // 50000*64*4 = 12.8 MB scratch

    gcn_init_bias<<<(N_NODES * F + 255) / 256, 256, 0, stream>>>(out, bias);
    gcn_gemm_wmma<<<(ROW_TILES + TILES_PER_BLOCK - 1) / TILES_PER_BLOCK, 256, 0, stream>>>(feat, W, H);
    gcn_scatter<<<(N_EDGES + 7) / 8, 256, 0, stream>>>(rows, cols, vals, H, out);
}